// SPDTangentSpace_38199439131090
// MI455X (gfx1250) — compile-verified
//
#include <hip/hip_runtime.h>
#include <math.h>

typedef __attribute__((ext_vector_type(2))) float v2f;
typedef __attribute__((ext_vector_type(8))) float v8f;

#define NDIM   256
#define NN     (NDIM * NDIM)           // 65536 elements per matrix
#define BATCH  512
#define THREADS 256                    // 8 wave32 per block
#define TRI    (NDIM * (NDIM + 1) / 2) // 32896
#define ASTR   34                      // padded LDS stride (floats) for A panel (128x32)
#define BSTR2  144                     // padded LDS stride (float2) for paired B panel
                                       // 144*2 = 288 = 32 mod 64 -> halves hit disjoint banks

// ---------------------------------------------------------------------------
// Workgroup-wide 256x256x256 f32 GEMM:  C = alpha*(A*B) + beta*I
// V_WMMA_F32_16X16X4_F32. 8 waves arranged 2x4, each wave computes a 64x32
// block (4x2 tiles of 16x16) of each 128x128 quadrant of C. K is staged in
// 32-wide chunks through LDS with a register-buffered software pipeline so
// global loads for chunk k+1 overlap the WMMA work of chunk k.
// B is stored in LDS in fragment-ready pair layout: (B[2p][n], B[2p+1][n])
// adjacent, so each B fragment is one aligned, bank-conflict-free ds_load_b64.
// ---------------------------------------------------------------------------
__device__ __forceinline__ void gemm256(const float* __restrict__ Ag,
                                        const float* __restrict__ Bg,
                                        float* __restrict__ Cg,
                                        float alpha, float beta,
                                        float* ldsA, float* ldsB)
{
    const int tid  = threadIdx.x;
    const int lane = tid & 31;
    const int wave = tid >> 5;
    const int wr   = wave >> 2;      // 0..1 : wave row within quadrant
    const int wc   = wave & 3;       // 0..3 : wave col within quadrant
    const int half = lane >> 4;      // 0/1  : lane half (K / M-half select)
    const int l16  = lane & 15;

    // fixed per-thread staging coordinates
    const int ar[4] = { (0 * THREADS + tid) >> 3, (1 * THREADS + tid) >> 3,
                        (2 * THREADS + tid) >> 3, (3 * THREADS + tid) >> 3 };
    const int ac    = (tid & 7) << 2;              // same for all i (tid pattern)
    const int br[4] = { (0 * THREADS + tid) >> 5, (1 * THREADS + tid) >> 5,
                        (2 * THREADS + tid) >> 5, (3 * THREADS + tid) >> 5 };
    const int bc    = (tid & 31) << 2;

    for (int mi = 0; mi < 2; ++mi) {
        for (int ni = 0; ni < 2; ++ni) {
            v8f acc[4][2];
#pragma unroll
            for (int tr = 0; tr < 4; ++tr)
#pragma unroll
                for (int tc = 0; tc < 2; ++tc)
                    acc[tr][tc] = v8f{0.f, 0.f, 0.f, 0.f, 0.f, 0.f, 0.f, 0.f};

            float4 aR[4], bR[4];
            // ---- pipeline prologue: load chunk 0 into registers ----
#pragma unroll
            for (int i = 0; i < 4; ++i)
                aR[i] = *(const float4*)&Ag[(size_t)(mi * 128 + ar[i]) * NDIM + 0 + ac];
#pragma unroll
            for (int i = 0; i < 4; ++i)
                bR[i] = *(const float4*)&Bg[(size_t)(0 + br[i]) * NDIM + ni * 128 + bc];

            for (int kc = 0; kc < 8; ++kc) {
                __syncthreads();     // previous chunk's compute done reading LDS
                // ---- commit registers -> LDS ----
#pragma unroll
                for (int i = 0; i < 4; ++i) {
                    float* d = &ldsA[ar[i] * ASTR + ac];
                    d[0] = aR[i].x; d[1] = aR[i].y; d[2] = aR[i].z; d[3] = aR[i].w;
                }
#pragma unroll
                for (int i = 0; i < 4; ++i) {
                    // paired layout: float2 (B[2p][n], B[2p+1][n])
                    float* d = &ldsB[((br[i] >> 1) * BSTR2 + bc) * 2 + (br[i] & 1)];
                    d[0] = bR[i].x; d[2] = bR[i].y; d[4] = bR[i].z; d[6] = bR[i].w;
                }
                // ---- issue global loads for next chunk (overlap with compute) ----
                if (kc < 7) {
                    const int kk1 = (kc + 1) * 32;
#pragma unroll
                    for (int i = 0; i < 4; ++i)
                        aR[i] = *(const float4*)&Ag[(size_t)(mi * 128 + ar[i]) * NDIM + kk1 + ac];
#pragma unroll
                    for (int i = 0; i < 4; ++i)
                        bR[i] = *(const float4*)&Bg[(size_t)(kk1 + br[i]) * NDIM + ni * 128 + bc];
                }
                __syncthreads();     // LDS panels ready

                // ---- 64 WMMAs on this K chunk ----
#pragma unroll
                for (int k4 = 0; k4 < 32; k4 += 4) {
                    v2f bf[2];
#pragma unroll
                    for (int tc = 0; tc < 2; ++tc) {
                        const int nloc = wc * 32 + tc * 16 + l16;
                        bf[tc] = *(const v2f*)&ldsB[(((k4 >> 1) + half) * BSTR2 + nloc) * 2];
                    }
#pragma unroll
                    for (int tr = 0; tr < 4; ++tr) {
                        const int row = wr * 64 + tr * 16 + l16;
                        v2f af = *(const v2f*)&ldsA[row * ASTR + k4 + 2 * half];
#pragma unroll
                        for (int tc = 0; tc < 2; ++tc) {
                            acc[tr][tc] = __builtin_amdgcn_wmma_f32_16x16x4_f32(
                                false, af, false, bf[tc], (short)0,
                                acc[tr][tc], false, false);
                        }
                    }
                }
            }
            // ---- epilogue: C = alpha*acc + beta*I ----
#pragma unroll
            for (int tr = 0; tr < 4; ++tr)
#pragma unroll
                for (int tc = 0; tc < 2; ++tc)
#pragma unroll
                    for (int v = 0; v < 8; ++v) {
                        int r = mi * 128 + wr * 64 + tr * 16 + half * 8 + v;
                        int c = ni * 128 + wc * 32 + tc * 16 + l16;
                        float val = alpha * acc[tr][tc][v] + ((r == c) ? beta : 0.f);
                        Cg[(size_t)r * NDIM + c] = val;
                    }
        }
    }
    __syncthreads();
}

// ---------------------------------------------------------------------------
// Persistent kernel: each block owns matrices mat = blockIdx.x, +nblk, ...
// and a private 5-matrix workspace slice.
// ---------------------------------------------------------------------------
__global__ __launch_bounds__(THREADS)
void spd_logm_kernel(const float* __restrict__ X, float* __restrict__ out,
                     float* __restrict__ ws, int nblk)
{
    __shared__ float ldsA[128 * ASTR];          // 17408 B
    __shared__ float ldsB[16 * BSTR2 * 2];      // 18432 B (paired layout)
    __shared__ float red[THREADS];
    __shared__ float s_inv, s_logc;

    const int tid = threadIdx.x;
    float* base = ws + (size_t)blockIdx.x * 5 * NN;

    for (int mat = blockIdx.x; mat < BATCH; mat += nblk) {
        const float* Ain = X + (size_t)mat * NN;
        float *Y  = base;
        float *Z  = base + (size_t)NN;
        float *W  = base + (size_t)2 * NN;
        float *T1 = base + (size_t)3 * NN;
        float *T2 = base + (size_t)4 * NN;

        // ---- c = ||A||_F (>= lambda_max for SPD) ----
        float s = 0.f;
        for (int i = tid; i < NN; i += THREADS) { float v = Ain[i]; s += v * v; }
        red[tid] = s; __syncthreads();
        for (int off = THREADS / 2; off > 0; off >>= 1) {
            if (tid < off) red[tid] += red[tid + off];
            __syncthreads();
        }
        if (tid == 0) {
            float c = sqrtf(red[0]);
            s_inv  = 1.f / c;
            s_logc = logf(c);
        }
        __syncthreads();
        const float inv  = s_inv;
        const float logc = s_logc;

        // ---- Y = A/c (spectrum in (0,1]), Z = I ----
        for (int i = tid; i < NN; i += THREADS) {
            Y[i] = Ain[i] * inv;
            Z[i] = ((i >> 8) == (i & 255)) ? 1.f : 0.f;
        }
        __syncthreads();

        // ---- 8 levels of Newton-Schulz matrix square root ----
        static constexpr int kIters[8] = {20, 13, 10, 8, 7, 6, 5, 5};
        for (int lvl = 0; lvl < 8; ++lvl) {
            if (lvl) {
                for (int i = tid; i < NN; i += THREADS)
                    Z[i] = ((i >> 8) == (i & 255)) ? 1.f : 0.f;
                __syncthreads();
            }
            for (int it = 0; it < kIters[lvl]; ++it) {
                gemm256(Z, Y, W,  -0.5f, 1.5f, ldsA, ldsB);  // W = 1.5I - 0.5*Z*Y
                gemm256(Y, W, T1,  1.f,  0.f,  ldsA, ldsB);  // Ynew = Y*W
                gemm256(W, Z, T2,  1.f,  0.f,  ldsA, ldsB);  // Znew = W*Z
                float* t = Y; Y = T1; T1 = t;
                t = Z; Z = T2; T2 = t;
            }
        }

        // ---- E = S - I (into W), Horner seed M = (1/7) I (into T1) ----
        for (int i = tid; i < NN; i += THREADS) {
            bool diag = ((i >> 8) == (i & 255));
            W[i]  = Y[i] - (diag ? 1.f : 0.f);
            T1[i] = diag ? (1.f / 7.f) : 0.f;
        }
        __syncthreads();

        // ---- Mercator series via Horner: q = c1 + c2 E + ... + c7 E^6 ----
        float* M  = T1;
        float* Tn = T2;
        for (int j = 6; j >= 1; --j) {
            float cj = ((j & 1) ? 1.f : -1.f) / (float)j;
            gemm256(W, M, Tn, 1.f, cj, ldsA, ldsB);          // M = cj*I + E*M
            float* t = M; M = Tn; Tn = t;
        }
        // ---- logm = 2^8 * E*q(E) + log(c)*I   (into Z, no longer needed) ----
        gemm256(W, M, Z, 256.f, logc, ldsA, ldsB);

        // ---- vectorize upper triangle (row-major triu, incl. diagonal) ----
        for (int lin = tid; lin < NN; lin += THREADS) {
            int r = lin >> 8, c = lin & 255;
            if (c >= r) {
                int idx = r * NDIM - ((r * (r - 1)) >> 1) + (c - r);
                out[(size_t)mat * TRI + idx] = Z[lin];
            }
        }
        __syncthreads();
    }
}

extern "C" void kernel_launch(void* const* d_in, const int* in_sizes, int n_in,
                              void* d_out, int out_size, void* d_ws, size_t ws_size,
                              hipStream_t stream)
{
    const float* X = (const float*)d_in[0];
    float* out = (float*)d_out;
    float* ws  = (float*)d_ws;

    // Each persistent block needs 5 scratch matrices of 256 KB.
    const size_t per_blk = (size_t)5 * NN * sizeof(float);
    int nblk = (int)(ws_size / per_blk);
    if (nblk < 1) nblk = 1;
    if (nblk > BATCH) nblk = BATCH;

    spd_logm_kernel<<<nblk, THREADS, 0, stream>>>(X, out, ws, nblk);
}